// PointPillarsEncoder_65274912965084
// MI455X (gfx1250) — compile-verified
//
#include <hip/hip_runtime.h>
#include <hip/hip_bf16.h>
#include <stdint.h>

// ---------------- constants (match reference) ----------------
#define XMIN  (-40.0f)
#define YMIN  (-40.0f)
#define ZMINC (-3.0f)
#define ZMAXC (3.0f)
#define HGRID 256
#define WGRID 256
#define TCAP  20
#define COUT  32
#define FIN   10
#define FPAD  12
#define VXSZ  (80.0f / 256.0f)
#define VYSZ  (80.0f / 256.0f)
#define ZCEN  0.0f
#define BNEPS 1e-5f

typedef float v2f __attribute__((ext_vector_type(2)));
typedef float v8f __attribute__((ext_vector_type(8)));

// monotone float <-> sortable-uint encoding (so we can use integer atomicMax/Min)
__device__ __forceinline__ uint32_t fkey(float f) {
    uint32_t u = __float_as_uint(f);
    return (u & 0x80000000u) ? ~u : (u | 0x80000000u);
}
__device__ __forceinline__ float fdec(uint32_t k) {
    uint32_t u = (k & 0x80000000u) ? (k & 0x7FFFFFFFu) : ~k;
    return __uint_as_float(u);
}

// ---------------- kernel 1: init workspace ----------------
__global__ void pp_init(uint32_t* maxkey, uint32_t* minkey, uint32_t* cnt,
                        float* gsum, float* gsumsq, uint32_t* occ,
                        int nkeys, int ptot) {
    int t = blockIdx.x * blockDim.x + threadIdx.x;
    if (t < nkeys) { maxkey[t] = 0u; minkey[t] = 0xFFFFFFFFu; }
    if (t < ptot)  cnt[t] = 0u;
    if (t < COUT)  { gsum[t] = 0.0f; gsumsq[t] = 0.0f; }
    if (t == 0)    *occ = 0u;
}

// ---------------- kernel 2: per-point PFN linear via f32 WMMA + scatter ----------------
// 256 threads = 8 waves; each wave handles 32 points (one per lane).
// A: 16x4 f32 tile (2 VGPRs), B: 4x16 (2 VGPRs), C/D: 16x16 f32 (8 VGPRs).
__global__ void __launch_bounds__(256)
pp_points(const float* __restrict__ pts, const float* __restrict__ wgt,
          uint32_t* maxkey, uint32_t* minkey, uint32_t* cnt,
          float* gsum, float* gsumsq, uint32_t* occ,
          int totalPts, int nPerB, int ptot) {
    __shared__ float ldsF[8][32][FPAD];
    __shared__ int   ldsPid[8][32];
    __shared__ float lsum[COUT], lsumsq[COUT];

    const int tid  = threadIdx.x;
    const int wv   = tid >> 5;
    const int lane = tid & 31;
    const int half = lane >> 4;    // 0 or 1
    const int l16  = lane & 15;

    if (tid < COUT) { lsum[tid] = 0.0f; lsumsq[tid] = 0.0f; }

    // ---- load this lane's point, compute pillar id + features ----
    const int g = blockIdx.x * 256 + tid;
    float r = 0.f, gg = 0.f, b = 0.f, x = 0.f, y = 0.f, z = 0.f;
    if (g < totalPts) {
        const float* p = pts + (size_t)g * 6;
        r = p[0]; gg = p[1]; b = p[2]; x = p[3]; y = p[4]; z = p[5];
    }
    const int ii = (int)truncf((x - XMIN) / VXSZ);
    const int jj = (int)truncf((y - YMIN) / VYSZ);
    const bool valid = (g < totalPts) && (ii >= 0) && (ii < WGRID) &&
                       (jj >= 0) && (jj < HGRID) && (z >= ZMINC) && (z < ZMAXC);
    const int bidx = (g < totalPts) ? (g / nPerB) : 0;
    const int pid  = valid ? (bidx * (HGRID * WGRID) + jj * WGRID + ii) : -1;

    const float cx = XMIN + ((float)ii + 0.5f) * VXSZ;
    const float cy = YMIN + ((float)jj + 0.5f) * VYSZ;

    float f[FPAD];
    f[0] = x; f[1] = y; f[2] = z; f[3] = r; f[4] = gg; f[5] = b;
    f[6] = x - cx; f[7] = y - cy; f[8] = z - ZCEN; f[9] = 1.0f;
    f[10] = 0.0f; f[11] = 0.0f;

    #pragma unroll
    for (int k = 0; k < FPAD; ++k) ldsF[wv][lane][k] = f[k];
    ldsPid[wv][lane] = pid;

    if (valid) {
        uint32_t old = atomicAdd(&cnt[pid], 1u);
        if (old == 0u) atomicAdd(occ, 1u);
    }

    // ---- B fragments from weight (B[k][c] = W[c][k]); 3 K-steps x 2 N-tiles ----
    v2f bf[3][2];
    #pragma unroll
    for (int s = 0; s < 3; ++s) {
        #pragma unroll
        for (int nt = 0; nt < 2; ++nt) {
            const int c  = nt * 16 + l16;
            const int k0 = 4 * s + half * 2;
            bf[s][nt][0] = (k0     < FIN) ? wgt[c * FIN + k0]     : 0.0f;
            bf[s][nt][1] = (k0 + 1 < FIN) ? wgt[c * FIN + k0 + 1] : 0.0f;
        }
    }

    __syncthreads();

    // ---- WMMA: lin(32 pts x 32 ch) = A(32x12) * B(12x32), fp32 ----
    v8f acc[2][2] = {};
    #pragma unroll
    for (int s = 0; s < 3; ++s) {
        const int kk = 4 * s + half * 2;
        #pragma unroll
        for (int mt = 0; mt < 2; ++mt) {
            const int m = mt * 16 + l16;
            v2f a;
            a[0] = ldsF[wv][m][kk];
            a[1] = ldsF[wv][m][kk + 1];
            #pragma unroll
            for (int nt = 0; nt < 2; ++nt) {
                acc[mt][nt] = __builtin_amdgcn_wmma_f32_16x16x4_f32(
                    false, a, false, bf[s][nt], (short)0, acc[mt][nt], false, false);
            }
        }
    }

    // ---- scatter: per-pillar max/min keys + channel sums ----
    #pragma unroll
    for (int nt = 0; nt < 2; ++nt) {
        const int c = nt * 16 + l16;
        float s1 = 0.0f, s2 = 0.0f;
        #pragma unroll
        for (int mt = 0; mt < 2; ++mt) {
            #pragma unroll
            for (int rr = 0; rr < 8; ++rr) {
                const int m    = mt * 16 + rr + half * 8;
                const int pidm = ldsPid[wv][m];
                const float v  = acc[mt][nt][rr];
                if (pidm >= 0) {
                    const size_t idx = (size_t)c * (size_t)ptot + (size_t)pidm;
                    const uint32_t k = fkey(v);
                    atomicMax(&maxkey[idx], k);
                    atomicMin(&minkey[idx], k);
                    s1 += v; s2 += v * v;
                }
            }
        }
        atomicAdd(&lsum[c], s1);
        atomicAdd(&lsumsq[c], s2);
    }

    __syncthreads();
    if (tid < COUT) {
        atomicAdd(&gsum[tid],   lsum[tid]);
        atomicAdd(&gsumsq[tid], lsumsq[tid]);
    }
}

// ---------------- kernel 3: BN batch stats -> per-channel affine (a, b) ----------------
__global__ void pp_stats(const float* gsum, const float* gsumsq, const uint32_t* occ,
                         const float* gamma, const float* beta, float* ab) {
    int c = threadIdx.x;
    if (c >= COUT) return;
    float n = (float)(*occ) * (float)TCAP;
    n = fmaxf(n, 1.0f);
    const float mean = gsum[c] / n;
    float var = gsumsq[c] / n - mean * mean;
    var = fmaxf(var, 0.0f);
    const float a = gamma[c] * rsqrtf(var + BNEPS);
    ab[c]        = a;
    ab[COUT + c] = beta[c] - mean * a;
}

// ---------------- kernel 4: finalize BEV (B, C, H, W) ----------------
__global__ void pp_final(const uint32_t* __restrict__ maxkey,
                         const uint32_t* __restrict__ minkey,
                         const uint32_t* __restrict__ cnt,
                         const float* __restrict__ ab,
                         float* __restrict__ out, int total, int ptot) {
    const int t = blockIdx.x * blockDim.x + threadIdx.x;
    if (t >= total) return;
    const int HW = HGRID * WGRID;
    const int hw = t % HW;
    const int c  = (t / HW) % COUT;
    const int bb = t / (HW * COUT);
    const int p  = bb * HW + hw;

    float o = 0.0f;
    const uint32_t n = cnt[p];
    if (n > 0u) {
        const size_t idx = (size_t)c * (size_t)ptot + (size_t)p;
        float mx = fdec(maxkey[idx]);
        float mn = fdec(minkey[idx]);
        if (n < (uint32_t)TCAP) { mx = fmaxf(mx, 0.0f); mn = fminf(mn, 0.0f); }
        const float a  = ab[c];
        const float bc = ab[COUT + c];
        const float ch = (a >= 0.0f) ? mx : mn;
        o = fmaxf(a * ch + bc, 0.0f);
    }
    out[t] = o;
}

// ---------------- host launcher ----------------
extern "C" void kernel_launch(void* const* d_in, const int* in_sizes, int n_in,
                              void* d_out, int out_size, void* d_ws, size_t ws_size,
                              hipStream_t stream) {
    const float* pts   = (const float*)d_in[0];   // (B, N, 6)
    const float* wgt   = (const float*)d_in[1];   // (32, 10)
    const float* gamma = (const float*)d_in[2];   // (32)
    const float* beta  = (const float*)d_in[3];   // (32)
    float* out = (float*)d_out;                   // (B, 32, 256, 256)

    const int HW       = HGRID * WGRID;
    const int B        = out_size / (COUT * HW);  // = 2
    const int totalPts = in_sizes[0] / 6;         // = B*N
    const int nPerB    = totalPts / B;            // = N
    const int ptot     = B * HW;
    const int nkeys    = COUT * ptot;

    // workspace layout
    uint8_t* w = (uint8_t*)d_ws;
    uint32_t* maxkey = (uint32_t*)w;                         w += (size_t)nkeys * 4;
    uint32_t* minkey = (uint32_t*)w;                         w += (size_t)nkeys * 4;
    uint32_t* cnt    = (uint32_t*)w;                         w += (size_t)ptot * 4;
    float*    gsum   = (float*)w;                            w += COUT * 4;
    float*    gsumsq = (float*)w;                            w += COUT * 4;
    uint32_t* occ    = (uint32_t*)w;                         w += 64;  // padded
    float*    ab     = (float*)w;                            /* 2*COUT floats */

    const int ib = (nkeys + 255) / 256;
    pp_init<<<ib, 256, 0, stream>>>(maxkey, minkey, cnt, gsum, gsumsq, occ, nkeys, ptot);

    const int pb = (totalPts + 255) / 256;
    pp_points<<<pb, 256, 0, stream>>>(pts, wgt, maxkey, minkey, cnt,
                                      gsum, gsumsq, occ, totalPts, nPerB, ptot);

    pp_stats<<<1, 64, 0, stream>>>(gsum, gsumsq, occ, gamma, beta, ab);

    const int fb = (out_size + 255) / 256;
    pp_final<<<fb, 256, 0, stream>>>(maxkey, minkey, cnt, ab, out, out_size, ptot);
}